// deeper_gcn_model_81071802679784
// MI455X (gfx1250) — compile-verified
//
#include <hip/hip_runtime.h>
#include <hip/hip_bf16.h>

#define N_NODES 100000
#define E_EDGES 1600000
#define INP     128
#define EDGE_D  16
#define HID     64
#define L_LAYERS 3
#define C_CLS   10
#define G_GRAPHS 128
#define MSG_EPS 1e-7f
#define BN_EPS  1e-5f

// padded LDS row strides (floats) -> bank-conflict-free fragment reads,
// and every async-copy chunk stays b64/b128 aligned
#define EF_STRIDE 18    // 16-wide tile rows, 72 B (8B multiple)
#define XT_STRIDE 68    // 64-wide tile rows, 272 B (16B multiple)
#define PR_STRIDE 132   // 128-wide tile rows, 528 B (16B multiple)

typedef __attribute__((ext_vector_type(2))) float v2f;
typedef __attribute__((ext_vector_type(8))) float v8f;

// ---------------------------------------------------------------------------
// gfx1250 async global->LDS copies (ASYNCcnt-tracked Tensor/async path)
// ---------------------------------------------------------------------------
__device__ __forceinline__ void async_load_b64(unsigned lds_off, const void* g) {
    asm volatile("global_load_async_to_lds_b64 %0, %1, off"
                 :: "v"(lds_off), "v"(g) : "memory");
}
__device__ __forceinline__ void async_load_b128(unsigned lds_off, const void* g) {
    asm volatile("global_load_async_to_lds_b128 %0, %1, off"
                 :: "v"(lds_off), "v"(g) : "memory");
}
__device__ __forceinline__ void wait_async_all() {
    asm volatile("s_wait_asynccnt 0" ::: "memory");
}
__device__ __forceinline__ unsigned lds_addr(const void* p) {
    // generic LDS pointer = {SHARED_BASE, lds_offset} -> low 32 bits are the offset
    return (unsigned)(size_t)p;
}

// ---------------------------------------------------------------------------
// Utility: zero a float buffer
// ---------------------------------------------------------------------------
__global__ void zero_kernel(float* __restrict__ p, size_t n) {
    size_t i = (size_t)blockIdx.x * blockDim.x + threadIdx.x;
    if (i < n) p[i] = 0.0f;
}

// ---------------------------------------------------------------------------
// Pre-encoder: h = node_feats @ W_pre + b_pre      [N,128] x [128,64]
// A tile (16x128, 8KB) async-staged to LDS once per block; 4 waves each own a
// 16-col WMMA tile.  A-frag per ISA 16x4 f32 layout (lanes 0-15: K=k,k+1;
// lanes 16-31: K=k+2,k+3); C/D: VGPR v holds rows v / v+8.
// ---------------------------------------------------------------------------
__global__ __launch_bounds__(128) void gemm_pre_kernel(
    const float* __restrict__ A,   // [N, INP]
    const float* __restrict__ W,   // [INP, HID]
    const float* __restrict__ b,   // [HID]
    float* __restrict__ H)         // [N, HID]
{
    __shared__ __align__(16) float s_a[16 * PR_STRIDE];
    const int t = threadIdx.x;
    const int rowbase = blockIdx.x * 16;
    {   // 2048 floats: 128 lanes x 16 floats = 4 x b128 per lane, coalesced
        const int row = t >> 3, c0 = (t & 7) * 16;
#pragma unroll
        for (int j = 0; j < 4; ++j) {
            async_load_b128(lds_addr(&s_a[row * PR_STRIDE + c0 + j * 4]),
                            A + (size_t)(rowbase + row) * INP + c0 + j * 4);
        }
    }
    wait_async_all();
    __syncthreads();

    const int lane = t & 31, wave = t >> 5;
    const int l15 = lane & 15, hi = lane >> 4;
    const int bcol = wave * 16 + l15;

    v8f acc = {};
#pragma unroll
    for (int k = 0; k < INP; k += 4) {
        const int kk = k + hi * 2;
        v2f a;
        a.x = s_a[l15 * PR_STRIDE + kk];
        a.y = s_a[l15 * PR_STRIDE + kk + 1];
        v2f bb;
        bb.x = W[(size_t)kk * HID + bcol];
        bb.y = W[(size_t)(kk + 1) * HID + bcol];
        acc = __builtin_amdgcn_wmma_f32_16x16x4_f32(
            false, a, false, bb, (short)0, acc, false, false);
    }
    const float bias = b[bcol];
#pragma unroll
    for (int v = 0; v < 8; ++v) {
        const int row = rowbase + v + hi * 8;
        H[(size_t)row * HID + bcol] = acc[v] + bias;
    }
}

// ---------------------------------------------------------------------------
// Edge pass 1: he = edge_feats @ We + be ; m = relu(h[src]+he)+eps
// atomicMax the per-(dst,channel) max (positive floats order as ints).
// Edge-feature tile (16x16, 1KB) async-staged to LDS, shared by all 4 waves.
// ---------------------------------------------------------------------------
__global__ __launch_bounds__(128) void edge_pass1_kernel(
    const float* __restrict__ EF,   // [E, 16]
    const float* __restrict__ We,   // [16, 64]   (layer slice)
    const float* __restrict__ be,   // [64]
    const float* __restrict__ H,    // [N, 64]
    const int*   __restrict__ src,
    const int*   __restrict__ dst,
    float* __restrict__ MX)         // [N, 64]  (zero-initialized)
{
    __shared__ __align__(16) float s_ef[16 * EF_STRIDE];
    __shared__ int s_sd[32];
    const int t = threadIdx.x;
    const int ebase = blockIdx.x * 16;
    {   // 256 floats: 128 lanes x 2 floats = 1 x b64 per lane, coalesced
        const int row = t >> 3, c0 = (t & 7) * 2;
        async_load_b64(lds_addr(&s_ef[row * EF_STRIDE + c0]),
                       EF + (size_t)(ebase + row) * EDGE_D + c0);
    }
    if (t < 32) s_sd[t] = (t < 16) ? src[ebase + t] : dst[ebase + t - 16];
    wait_async_all();
    __syncthreads();

    const int lane = t & 31, wave = t >> 5;
    const int l15 = lane & 15, hi = lane >> 4;
    const int bcol = wave * 16 + l15;

    v8f acc = {};
#pragma unroll
    for (int k = 0; k < EDGE_D; k += 4) {
        const int kk = k + hi * 2;
        v2f a;
        a.x = s_ef[l15 * EF_STRIDE + kk];
        a.y = s_ef[l15 * EF_STRIDE + kk + 1];
        v2f bb;
        bb.x = We[kk * HID + bcol];
        bb.y = We[(kk + 1) * HID + bcol];
        acc = __builtin_amdgcn_wmma_f32_16x16x4_f32(
            false, a, false, bb, (short)0, acc, false, false);
    }
    const float bias = be[bcol];
#pragma unroll
    for (int v = 0; v < 8; ++v) {
        const int er = v + hi * 8;
        const int s  = s_sd[er];
        const int d  = s_sd[16 + er];
        float mval = fmaxf(H[(size_t)s * HID + bcol] + acc[v] + bias, 0.0f) + MSG_EPS;
        atomicMax((int*)&MX[(size_t)d * HID + bcol], __float_as_int(mval));
    }
}

// ---------------------------------------------------------------------------
// Edge pass 2: recompute m (bit-identical: same code, same inputs), then
// ex = exp(m - mx[dst]);  sden += ex ; snum += m*ex.
// Avoids materializing m[E,64] (saves ~820 MB HBM traffic per layer; the
// whole per-layer working set is ~L2-resident on the 192 MB L2).
// ---------------------------------------------------------------------------
__global__ __launch_bounds__(128) void edge_pass2_kernel(
    const float* __restrict__ EF,
    const float* __restrict__ We,
    const float* __restrict__ be,
    const float* __restrict__ H,
    const int*   __restrict__ src,
    const int*   __restrict__ dst,
    const float* __restrict__ MX,
    float* __restrict__ sden,       // [N,64] zeroed
    float* __restrict__ snum)       // [N,64] zeroed
{
    __shared__ __align__(16) float s_ef[16 * EF_STRIDE];
    __shared__ int s_sd[32];
    const int t = threadIdx.x;
    const int ebase = blockIdx.x * 16;
    {
        const int row = t >> 3, c0 = (t & 7) * 2;
        async_load_b64(lds_addr(&s_ef[row * EF_STRIDE + c0]),
                       EF + (size_t)(ebase + row) * EDGE_D + c0);
    }
    if (t < 32) s_sd[t] = (t < 16) ? src[ebase + t] : dst[ebase + t - 16];
    wait_async_all();
    __syncthreads();

    const int lane = t & 31, wave = t >> 5;
    const int l15 = lane & 15, hi = lane >> 4;
    const int bcol = wave * 16 + l15;

    v8f acc = {};
#pragma unroll
    for (int k = 0; k < EDGE_D; k += 4) {
        const int kk = k + hi * 2;
        v2f a;
        a.x = s_ef[l15 * EF_STRIDE + kk];
        a.y = s_ef[l15 * EF_STRIDE + kk + 1];
        v2f bb;
        bb.x = We[kk * HID + bcol];
        bb.y = We[(kk + 1) * HID + bcol];
        acc = __builtin_amdgcn_wmma_f32_16x16x4_f32(
            false, a, false, bb, (short)0, acc, false, false);
    }
    const float bias = be[bcol];
#pragma unroll
    for (int v = 0; v < 8; ++v) {
        const int er = v + hi * 8;
        const int s  = s_sd[er];
        const int d  = s_sd[16 + er];
        const size_t di = (size_t)d * HID + bcol;
        float mval = fmaxf(H[(size_t)s * HID + bcol] + acc[v] + bias, 0.0f) + MSG_EPS;
        float ex = __expf(mval - MX[di]);
        atomicAdd(&sden[di], ex);
        atomicAdd(&snum[di], mval * ex);
    }
}

// ---------------------------------------------------------------------------
// x = h + agg,  agg = snum / (sden + 1e-16)
// ---------------------------------------------------------------------------
__global__ void node_update_kernel(
    const float* __restrict__ H,
    const float* __restrict__ snum,
    const float* __restrict__ sden,
    float* __restrict__ X)
{
    const size_t i = (size_t)blockIdx.x * blockDim.x + threadIdx.x;  // N*64 exact
    X[i] = H[i] + snum[i] / (sden[i] + 1e-16f);
}

// ---------------------------------------------------------------------------
// hv = x @ Wm + bm + h     [N,64] x [64,64], + residual
// A tile (16x64, 4KB) async-staged to LDS once per block.
// ---------------------------------------------------------------------------
__global__ __launch_bounds__(128) void gemm_node_kernel(
    const float* __restrict__ X,    // [N, 64]
    const float* __restrict__ Wm,   // [64, 64]  (layer slice)
    const float* __restrict__ bm,   // [64]
    const float* __restrict__ H,    // [N, 64]   residual
    float* __restrict__ HV)         // [N, 64]
{
    __shared__ __align__(16) float s_x[16 * XT_STRIDE];
    const int t = threadIdx.x;
    const int rowbase = blockIdx.x * 16;
    {   // 1024 floats: 128 lanes x 8 floats = 2 x b128 per lane, coalesced
        const int row = t >> 3, c0 = (t & 7) * 8;
#pragma unroll
        for (int j = 0; j < 2; ++j) {
            async_load_b128(lds_addr(&s_x[row * XT_STRIDE + c0 + j * 4]),
                            X + (size_t)(rowbase + row) * HID + c0 + j * 4);
        }
    }
    wait_async_all();
    __syncthreads();

    const int lane = t & 31, wave = t >> 5;
    const int l15 = lane & 15, hi = lane >> 4;
    const int bcol = wave * 16 + l15;

    v8f acc = {};
#pragma unroll
    for (int k = 0; k < HID; k += 4) {
        const int kk = k + hi * 2;
        v2f a;
        a.x = s_x[l15 * XT_STRIDE + kk];
        a.y = s_x[l15 * XT_STRIDE + kk + 1];
        v2f bb;
        bb.x = Wm[kk * HID + bcol];
        bb.y = Wm[(kk + 1) * HID + bcol];
        acc = __builtin_amdgcn_wmma_f32_16x16x4_f32(
            false, a, false, bb, (short)0, acc, false, false);
    }
    const float bias = bm[bcol];
#pragma unroll
    for (int v = 0; v < 8; ++v) {
        const int row = rowbase + v + hi * 8;
        HV[(size_t)row * HID + bcol] = acc[v] + bias + H[(size_t)row * HID + bcol];
    }
}

// ---------------------------------------------------------------------------
// BatchNorm statistics: per-channel sum/sumsq; LDS reduce then 64 atomics.
// ---------------------------------------------------------------------------
__global__ __launch_bounds__(256) void bn_stats_kernel(
    const float* __restrict__ HV,
    float* __restrict__ stats,     // [0..63]=sum, [64..127]=sumsq (zeroed)
    int n)
{
    __shared__ float s_sum[256];
    __shared__ float s_sq[256];
    const int c  = threadIdx.x & 63;
    const int rg = threadIdx.x >> 6;       // 0..3
    const int row0 = blockIdx.x * 128;
    float s = 0.0f, q = 0.0f;
    for (int r = row0 + rg; r < row0 + 128 && r < n; r += 4) {
        const float v = HV[(size_t)r * HID + c];
        s += v;
        q += v * v;
    }
    s_sum[threadIdx.x] = s;
    s_sq[threadIdx.x]  = q;
    __syncthreads();
    if (rg == 0) {
        s = s_sum[c] + s_sum[c + 64] + s_sum[c + 128] + s_sum[c + 192];
        q = s_sq[c]  + s_sq[c + 64]  + s_sq[c + 128]  + s_sq[c + 192];
        atomicAdd(&stats[c], s);
        atomicAdd(&stats[64 + c], q);
    }
}

__global__ void bn_finalize_kernel(
    float* __restrict__ stats,
    const float* __restrict__ gamma,
    const float* __restrict__ beta,
    float n)
{
    const int c = threadIdx.x;            // 64 threads
    const float mu  = stats[c] / n;
    const float var = stats[64 + c] / n - mu * mu;
    const float sc  = gamma[c] * rsqrtf(var + BN_EPS);
    stats[128 + c] = sc;
    stats[192 + c] = beta[c] - mu * sc;
}

__global__ void bn_apply_kernel(
    const float* __restrict__ HV,
    const float* __restrict__ stats,
    float* __restrict__ H)
{
    const size_t i = (size_t)blockIdx.x * blockDim.x + threadIdx.x;  // N*64 exact
    const int c = (int)(i & 63);
    H[i] = fmaxf(HV[i] * stats[128 + c] + stats[192 + c], 0.0f);
}

// ---------------------------------------------------------------------------
// Mean pooling per graph + classifier head
// ---------------------------------------------------------------------------
__global__ void pool_kernel(
    const float* __restrict__ H,
    const int*   __restrict__ gid,
    float* __restrict__ pooled,    // [G, 64] zeroed
    float* __restrict__ cnt)       // [G] zeroed
{
    const size_t i = (size_t)blockIdx.x * blockDim.x + threadIdx.x;  // N*64 exact
    const int n = (int)(i >> 6);
    const int c = (int)(i & 63);
    const int g = gid[n];
    atomicAdd(&pooled[(size_t)g * HID + c], H[i]);
    if (c == 0) atomicAdd(&cnt[g], 1.0f);
}

__global__ void head_kernel(
    const float* __restrict__ pooled,
    const float* __restrict__ cnt,
    const float* __restrict__ Wo,   // [64, 10]
    const float* __restrict__ bo,   // [10]
    float* __restrict__ out)        // [G, 10]
{
    const int idx = blockIdx.x * blockDim.x + threadIdx.x;
    if (idx >= G_GRAPHS * C_CLS) return;
    const int g = idx / C_CLS;
    const int j = idx % C_CLS;
    const float inv = 1.0f / fmaxf(cnt[g], 1.0f);
    float acc = bo[j];
#pragma unroll
    for (int c = 0; c < HID; ++c)
        acc += pooled[(size_t)g * HID + c] * inv * Wo[c * C_CLS + j];
    out[idx] = acc;
}

// ---------------------------------------------------------------------------
// Launch sequence
// ---------------------------------------------------------------------------
extern "C" void kernel_launch(void* const* d_in, const int* in_sizes, int n_in,
                              void* d_out, int out_size, void* d_ws, size_t ws_size,
                              hipStream_t stream) {
    const float* node_feats = (const float*)d_in[0];
    const float* edge_feats = (const float*)d_in[1];
    const int*   src        = (const int*)d_in[2];
    const int*   dst        = (const int*)d_in[3];
    const int*   gid        = (const int*)d_in[4];
    const float* W_pre      = (const float*)d_in[5];
    const float* b_pre      = (const float*)d_in[6];
    const float* We         = (const float*)d_in[7];
    const float* be         = (const float*)d_in[8];
    const float* Wm         = (const float*)d_in[9];
    const float* bm         = (const float*)d_in[10];
    const float* bn_gamma   = (const float*)d_in[11];
    const float* bn_beta    = (const float*)d_in[12];
    const float* W_out      = (const float*)d_in[13];
    const float* b_out      = (const float*)d_in[14];
    float* out = (float*)d_out;
    float* ws  = (float*)d_ws;

    const size_t NH = (size_t)N_NODES * HID;

    size_t o = 0;
    float* h      = ws + o; o += NH;
    float* x      = ws + o; o += NH;
    float* hv     = ws + o; o += NH;
    float* mx     = ws + o; o += NH;   // mx, sden, snum, stats contiguous
    float* sden   = ws + o; o += NH;
    float* snum   = ws + o; o += NH;
    float* stats  = ws + o; o += 256;
    float* pooled = ws + o; o += (size_t)G_GRAPHS * HID;  // pooled, cnt contiguous
    float* cnt    = ws + o; o += G_GRAPHS;

    // Pre-encoder (N divisible by 16 -> full tiles, EXEC all-ones for WMMA)
    gemm_pre_kernel<<<N_NODES / 16, 128, 0, stream>>>(node_feats, W_pre, b_pre, h);

    for (int l = 0; l < L_LAYERS; ++l) {
        const size_t zn = 3 * NH + 256;  // mx + sden + snum + stats
        zero_kernel<<<(unsigned)((zn + 255) / 256), 256, 0, stream>>>(mx, zn);

        edge_pass1_kernel<<<E_EDGES / 16, 128, 0, stream>>>(
            edge_feats, We + (size_t)l * EDGE_D * HID, be + l * HID,
            h, src, dst, mx);

        edge_pass2_kernel<<<E_EDGES / 16, 128, 0, stream>>>(
            edge_feats, We + (size_t)l * EDGE_D * HID, be + l * HID,
            h, src, dst, mx, sden, snum);

        node_update_kernel<<<(unsigned)(NH / 256), 256, 0, stream>>>(h, snum, sden, x);

        gemm_node_kernel<<<N_NODES / 16, 128, 0, stream>>>(
            x, Wm + (size_t)l * HID * HID, bm + l * HID, h, hv);

        bn_stats_kernel<<<(N_NODES + 127) / 128, 256, 0, stream>>>(hv, stats, N_NODES);
        bn_finalize_kernel<<<1, 64, 0, stream>>>(stats, bn_gamma + l * HID,
                                                 bn_beta + l * HID, (float)N_NODES);
        bn_apply_kernel<<<(unsigned)(NH / 256), 256, 0, stream>>>(hv, stats, h);
    }

    const size_t pz = (size_t)G_GRAPHS * HID + G_GRAPHS;
    zero_kernel<<<(unsigned)((pz + 255) / 256), 256, 0, stream>>>(pooled, pz);
    pool_kernel<<<(unsigned)(NH / 256), 256, 0, stream>>>(h, gid, pooled, cnt);
    head_kernel<<<(G_GRAPHS * C_CLS + 255) / 256, 256, 0, stream>>>(pooled, cnt, W_out, b_out, out);
}